// GPGmodel_without_NN_35330400976969
// MI455X (gfx1250) — compile-verified
//
#include <hip/hip_runtime.h>
#include <stdint.h>

#define N_NODES 262144      // B * NBUS
#define N_BUS   512
#define N_EDGE  2097152
#define N_LAYER 11          // 1 + NUM_LAYERS
#define TPB     256
#define EPT     4           // edges per thread per stage (16 bytes per array)
#define SC_GRID 512         // scatter blocks: each does E/(SC_GRID*TPB*EPT) = 4 chunks

// ---------------- CDNA5 async global->LDS helpers ----------------
__device__ __forceinline__ void async_b128(uint32_t lds_off, const void* g) {
  unsigned long long ga = (unsigned long long)(uintptr_t)g;
  asm volatile("global_load_async_to_lds_b128 %0, %1, off"
               :: "v"(lds_off), "v"(ga) : "memory");
}
__device__ __forceinline__ void wait_async_le3() {
  asm volatile("s_wait_asynccnt 0x3" ::: "memory");
}
__device__ __forceinline__ void wait_async_0() {
  asm volatile("s_wait_asynccnt 0x0" ::: "memory");
}

// ---------------- init: den = 100*diag(ybus), p = x0 - x1 ----------------
__global__ __launch_bounds__(TPB) void init_kernel(const float* __restrict__ x,
                                                   const float* __restrict__ ybus,
                                                   float* __restrict__ den,
                                                   float* __restrict__ p) {
  int n = blockIdx.x * blockDim.x + threadIdx.x;
  if (n >= N_NODES) return;
  int i = n & (N_BUS - 1);
  // diag element of graph b, bus i: ybus[b*NBUS*NBUS + i*NBUS + i] = ybus[n*NBUS + i]
  den[n] = 100.0f * ybus[(size_t)n * N_BUS + i];
  p[n]   = x[2 * n] - x[2 * n + 1];
}

// ---------------- zero accumulators + this layer's error slot ----------------
__global__ __launch_bounds__(TPB) void zero_kernel(float* __restrict__ aggr,
                                                   float* __restrict__ agg2,
                                                   float* __restrict__ err_slot) {
  int n = blockIdx.x * blockDim.x + threadIdx.x;
  if (n < N_NODES) { aggr[n] = 0.0f; agg2[n] = 0.0f; }
  if (n == 0) *err_slot = 0.0f;
}

// ---------------- scatter: aggr[dst] += theta[src] * (w*100) ----------------
// Edge streams (src,dst,w) are staged into LDS with double-buffered
// GLOBAL_LOAD_ASYNC_TO_LDS_B128; each thread consumes only its own 4-edge slot
// so no cross-thread barrier is needed, only s_wait_asynccnt (in-order done).
__global__ __launch_bounds__(TPB) void scatter_kernel(const int*  __restrict__ src,
                                                      const int*  __restrict__ dst,
                                                      const float* __restrict__ w,
                                                      const float* __restrict__ theta,
                                                      float* __restrict__ aggr) {
  __shared__ __align__(16) int   s_src[2][TPB * EPT];
  __shared__ __align__(16) int   s_dst[2][TPB * EPT];
  __shared__ __align__(16) float s_w  [2][TPB * EPT];

  const int t        = threadIdx.x;
  const int lane_elt = t * EPT;
  const long long chunk  = (long long)TPB * EPT;            // 1024 edges/block/iter
  const long long stride = (long long)gridDim.x * chunk;
  long long e = (long long)blockIdx.x * chunk;
  if (e >= N_EDGE) return;                                  // uniform per block

  // LDS byte offsets (low 32 bits of flat shared address == LDS address)
  uint32_t lsrc0 = (uint32_t)(uintptr_t)&s_src[0][lane_elt];
  uint32_t lsrc1 = (uint32_t)(uintptr_t)&s_src[1][lane_elt];
  uint32_t ldst0 = (uint32_t)(uintptr_t)&s_dst[0][lane_elt];
  uint32_t ldst1 = (uint32_t)(uintptr_t)&s_dst[1][lane_elt];
  uint32_t lw0   = (uint32_t)(uintptr_t)&s_w  [0][lane_elt];
  uint32_t lw1   = (uint32_t)(uintptr_t)&s_w  [1][lane_elt];

  { // prologue: stage chunk 0 into buffer 0
    long long ei = e + lane_elt;
    async_b128(lsrc0, src + ei);
    async_b128(ldst0, dst + ei);
    async_b128(lw0,   w   + ei);
  }
  int buf = 0;
  while (true) {
    long long enext = e + stride;
    bool more = enext < N_EDGE;                             // uniform per block
    if (more) {                                             // stage next chunk
      long long ei = enext + lane_elt;
      async_b128(buf ? lsrc0 : lsrc1, src + ei);
      async_b128(buf ? ldst0 : ldst1, dst + ei);
      async_b128(buf ? lw0   : lw1,   w   + ei);
      wait_async_le3();                                     // current chunk landed
    } else {
      wait_async_0();
    }
    const int4   sv = *(const int4*)  &s_src[buf][lane_elt];
    const int4   dv = *(const int4*)  &s_dst[buf][lane_elt];
    const float4 wv = *(const float4*)&s_w  [buf][lane_elt];
    atomicAdd(&aggr[dv.x], theta[sv.x] * (wv.x * 100.0f));
    atomicAdd(&aggr[dv.y], theta[sv.y] * (wv.y * 100.0f));
    atomicAdd(&aggr[dv.z], theta[sv.z] * (wv.z * 100.0f));
    atomicAdd(&aggr[dv.w], theta[sv.w] * (wv.w * 100.0f));
    if (!more) break;
    e = enext;
    buf ^= 1;
  }
}

// ---------------- update: out = where(den!=0, (p-aggr)/den - ref, 0) ----------------
__global__ __launch_bounds__(TPB) void update_kernel(const float* __restrict__ den,
                                                     const float* __restrict__ p,
                                                     const float* __restrict__ aggr,
                                                     float* __restrict__ out) {
  int n = blockIdx.x * blockDim.x + threadIdx.x;
  if (n >= N_NODES) return;
  int r = n & ~(N_BUS - 1);                 // reference bus (index 0) of this graph
  float dr  = den[r];
  float ref = (dr != 0.0f) ? (p[r] - aggr[r]) / dr : 0.0f;
  float d   = den[n];
  float o   = 0.0f;
  if (d != 0.0f) o = (p[n] - aggr[n]) / d - ref;
  out[n] = o;
}

// ---------------- reduce: err += sum |p - agg2| ----------------
__global__ __launch_bounds__(TPB) void reduce_kernel(const float* __restrict__ p,
                                                     const float* __restrict__ agg2,
                                                     float* __restrict__ err) {
  __shared__ float sm[TPB];
  float acc = 0.0f;
  for (int n = blockIdx.x * blockDim.x + threadIdx.x; n < N_NODES;
       n += gridDim.x * blockDim.x)
    acc += fabsf(p[n] - agg2[n]);
  sm[threadIdx.x] = acc;
  __syncthreads();
  for (int s = TPB / 2; s > 0; s >>= 1) {
    if (threadIdx.x < s) sm[threadIdx.x] += sm[threadIdx.x + s];
    __syncthreads();
  }
  if (threadIdx.x == 0) atomicAdd(err, sm[0]);
}

// ---------------- driver ----------------
extern "C" void kernel_launch(void* const* d_in, const int* in_sizes, int n_in,
                              void* d_out, int out_size, void* d_ws, size_t ws_size,
                              hipStream_t stream) {
  const float* x     = (const float*)d_in[0];
  // d_in[1] = y, only its shape matters (theta init = zeros)
  const float* ybus  = (const float*)d_in[2];
  const int*   ei_nd = (const int*)d_in[3];       // [2,E] no-diag
  const float* ea_nd = (const float*)d_in[4];     // [E]
  const int*   ei    = (const int*)d_in[5];       // [2,E] full
  const float* ea    = (const float*)d_in[6];     // [E]

  float* out  = (float*)d_out;                    // [N] theta, then [11] errors
  float* errs = out + N_NODES;

  float* den  = (float*)d_ws;                     // [N]
  float* p    = den  + N_NODES;                   // [N]
  float* aggr = p    + N_NODES;                   // [N]
  float* agg2 = aggr + N_NODES;                   // [N]  (4 MB total scratch)

  const int gN = N_NODES / TPB;                   // 1024 blocks for node-wide kernels

  init_kernel<<<gN, TPB, 0, stream>>>(x, ybus, den, p);

  for (int L = 0; L < N_LAYER; ++L) {
    zero_kernel<<<gN, TPB, 0, stream>>>(aggr, agg2, errs + L);
    if (L > 0)  // layer 0 has theta == 0, so its aggregation is identically zero
      scatter_kernel<<<SC_GRID, TPB, 0, stream>>>(ei_nd, ei_nd + N_EDGE, ea_nd,
                                                  out, aggr);
    update_kernel<<<gN, TPB, 0, stream>>>(den, p, aggr, out);
    scatter_kernel<<<SC_GRID, TPB, 0, stream>>>(ei, ei + N_EDGE, ea, out, agg2);
    reduce_kernel<<<256, TPB, 0, stream>>>(p, agg2, errs + L);
  }
}